// MyRNN_1795296329802
// MI455X (gfx1250) — compile-verified
//
#include <hip/hip_runtime.h>
#include <hip/hip_bf16.h>
#include <math.h>

// Problem constants (match reference)
#define BATCH 64
#define SEQ   512
#define DIN   512
#define HID   1024
#define LN_EPS 1e-3f

// LDS row pitch for h: even (8B-aligned b64 loads) and pitch%64==2 so the 16
// A-matrix lanes (stride = pitch words) hit 16 distinct LDS banks.
#define HPAD  1026

typedef float v2f __attribute__((ext_vector_type(2)));
typedef float v8f __attribute__((ext_vector_type(8)));

// ---------------------------------------------------------------------------
// Phase 1: xW = X @ W + b   (M = B*T = 32768, K = 512, N = 1024), fp32 WMMA.
// Grid: 2048 blocks (one 16-row M-tile each) x 512 threads (16 waves).
// Each wave computes a 16x64 C block (4 WMMA N-tiles), K-loop in chunks of 4.
// Result written straight into d_out (consumed in-place by phase 2).
// ---------------------------------------------------------------------------
__global__ __launch_bounds__(512)
void lnrnn_xw_gemm_f32(const float* __restrict__ X,
                       const float* __restrict__ W,
                       const float* __restrict__ bias,
                       float* __restrict__ out)
{
    const int lane = threadIdx.x & 31;
    const int wave = threadIdx.x >> 5;     // 0..15 -> N chunk of 64
    const int hi   = lane >> 4;            // half-wave select
    const int lm   = lane & 15;

    const int m0 = blockIdx.x * 16;        // 16 rows of M = B*T
    const int n0 = wave * 64;

    // A: lane holds M=lm, K = k0 + 2*hi + {0,1}  (ISA 16x4 f32 A layout)
    const float* __restrict__ arow = X + (size_t)(m0 + lm) * DIN;

    v8f acc[4] = {v8f{}, v8f{}, v8f{}, v8f{}};

    #pragma unroll 2
    for (int k0 = 0; k0 < DIN; k0 += 4) {
        v2f a = *(const v2f*)(arow + k0 + 2 * hi);
        #pragma unroll
        for (int tt = 0; tt < 4; ++tt) {
            const int n = n0 + tt * 16 + lm;
            // B: VGPR0 = rows k0 (lanes 0-15) / k0+2 (lanes 16-31),
            //    VGPR1 = rows k0+1 / k0+3
            v2f bm;
            bm.x = W[(size_t)(k0 + 2 * hi)     * HID + n];
            bm.y = W[(size_t)(k0 + 2 * hi + 1) * HID + n];
            acc[tt] = __builtin_amdgcn_wmma_f32_16x16x4_f32(
                false, a, false, bm, (short)0, acc[tt], false, false);
        }
    }

    // C/D layout: VGPR r -> row (r + 8*hi), col n0 + tt*16 + lm. Add bias, store.
    #pragma unroll
    for (int tt = 0; tt < 4; ++tt) {
        const int n = n0 + tt * 16 + lm;
        const float bn = bias[n];
        #pragma unroll
        for (int r = 0; r < 8; ++r) {
            const int m = m0 + r + 8 * hi;
            out[(size_t)m * HID + n] = acc[tt][r] + bn;
        }
    }
}

// ---------------------------------------------------------------------------
// Phase 2: recurrent scan. 4 workgroups, each owns 16 batch rows for the whole
// sequence; h[16 x 1024] lives in LDS, U streamed from L2 (4 MB, L2-resident).
// Per step: z = xW_t + h@U (fp32 WMMA), LayerNorm across H via cross-lane
// shuffles + LDS float atomics (ping-pong accumulators, 2 barriers/step),
// h_new = tanh(LN(z)) written back to LDS and to d_out in place of xW_t.
// ---------------------------------------------------------------------------
__global__ __launch_bounds__(512)
void lnrnn_recurrent_f32(const float* __restrict__ U,
                         const float* __restrict__ gamma,
                         const float* __restrict__ beta,
                         float* __restrict__ out)
{
    __shared__ float hbuf[16 * HPAD];      // hidden state, padded rows
    __shared__ float red[2][16][2];        // ping-pong [parity][row][sum,sumsq]

    const int lane = threadIdx.x & 31;
    const int wave = threadIdx.x >> 5;     // 0..15 -> N chunk of 64
    const int hi   = lane >> 4;
    const int lm   = lane & 15;
    const int b0   = blockIdx.x * 16;      // batch rows owned by this WG
    const int n0   = wave * 64;

    // Zero h0 and both reduction buffers.
    for (int i = threadIdx.x; i < 16 * HPAD; i += blockDim.x) hbuf[i] = 0.0f;
    if (threadIdx.x < 64) ((float*)red)[threadIdx.x] = 0.0f;

    // Per-lane LN affine params for this lane's 4 columns (gamma=1, beta=0 in
    // the reference init, but honor them).
    float gv[4], bv[4];
    #pragma unroll
    for (int tt = 0; tt < 4; ++tt) {
        const int n = n0 + tt * 16 + lm;
        gv[tt] = gamma[n];
        bv[tt] = beta[n];
    }
    __syncthreads();

    for (int t = 0; t < SEQ; ++t) {
        // Prefetch next timestep's xW rows into cache while we do the GEMM.
        if (t + 1 < SEQ) {
            __builtin_prefetch(
                &out[((size_t)(b0 + 8 * hi) * SEQ + (t + 1)) * HID + n0 + lm], 0, 1);
        }

        // ---- GEMM: acc = h @ U (16 x 1024 x 64 slice per wave) ----
        v8f acc[4] = {v8f{}, v8f{}, v8f{}, v8f{}};
        const float* __restrict__ hrow = &hbuf[lm * HPAD + 2 * hi];

        #pragma unroll 2
        for (int k0 = 0; k0 < HID; k0 += 4) {
            v2f a = *(const v2f*)(hrow + k0);   // ds_load_b64, bank-spread
            #pragma unroll
            for (int tt = 0; tt < 4; ++tt) {
                const int n = n0 + tt * 16 + lm;
                v2f bm;
                bm.x = U[(size_t)(k0 + 2 * hi)     * HID + n];
                bm.y = U[(size_t)(k0 + 2 * hi + 1) * HID + n];
                acc[tt] = __builtin_amdgcn_wmma_f32_16x16x4_f32(
                    false, a, false, bm, (short)0, acc[tt], false, false);
            }
        }

        // ---- z = xW_t + acc; per-lane partial LN statistics ----
        float psum[8], psq[8];
        #pragma unroll
        for (int r = 0; r < 8; ++r) { psum[r] = 0.0f; psq[r] = 0.0f; }

        #pragma unroll
        for (int tt = 0; tt < 4; ++tt) {
            const int n = n0 + tt * 16 + lm;
            #pragma unroll
            for (int r = 0; r < 8; ++r) {
                const int m = r + 8 * hi;
                const size_t idx = ((size_t)(b0 + m) * SEQ + t) * HID + n;
                const float z = out[idx] + acc[tt][r];
                acc[tt][r] = z;
                psum[r] += z;
                psq[r]  += z * z;
            }
        }

        // Reduce over the 16 lanes of each half-wave (xor masks stay in-half).
        #pragma unroll
        for (int off = 1; off < 16; off <<= 1) {
            #pragma unroll
            for (int r = 0; r < 8; ++r) {
                psum[r] += __shfl_xor(psum[r], off, 32);
                psq[r]  += __shfl_xor(psq[r],  off, 32);
            }
        }
        // One leader lane per half-wave accumulates into the step's buffer.
        if (lm == 0) {
            #pragma unroll
            for (int r = 0; r < 8; ++r) {
                atomicAdd(&red[t & 1][r + 8 * hi][0], psum[r]);  // ds_add_f32
                atomicAdd(&red[t & 1][r + 8 * hi][1], psq[r]);
            }
        }
        __syncthreads();   // all partials in; all GEMM reads of hbuf done

        // ---- LN + tanh; write h_new to LDS and (in place) to out ----
        #pragma unroll
        for (int r = 0; r < 8; ++r) {
            const int m = r + 8 * hi;
            const float s  = red[t & 1][m][0];
            const float sq = red[t & 1][m][1];
            const float mu  = s * (1.0f / HID);
            const float var = sq * (1.0f / HID) - mu * mu;
            const float rs  = rsqrtf(var + LN_EPS);
            #pragma unroll
            for (int tt = 0; tt < 4; ++tt) {
                const int n = n0 + tt * 16 + lm;
                const float hn = tanhf((acc[tt][r] - mu) * rs * gv[tt] + bv[tt]);
                hbuf[m * HPAD + n] = hn;
                out[((size_t)(b0 + m) * SEQ + t) * HID + n] = hn;
            }
        }
        // Re-zero the other parity's accumulators for step t+1 (safe: everyone
        // is past its last read, which happened before step t's first barrier).
        if (threadIdx.x < 32) ((float*)red[(t + 1) & 1])[threadIdx.x] = 0.0f;
        __syncthreads();   // h_new + zeroed accumulators visible to all waves
    }
}

extern "C" void kernel_launch(void* const* d_in, const int* in_sizes, int n_in,
                              void* d_out, int out_size, void* d_ws, size_t ws_size,
                              hipStream_t stream)
{
    (void)in_sizes; (void)n_in; (void)out_size; (void)d_ws; (void)ws_size;
    const float* X     = (const float*)d_in[0];  // [64, 512, 512]
    const float* W     = (const float*)d_in[1];  // [512, 1024]
    const float* U     = (const float*)d_in[2];  // [1024, 1024]
    const float* bias  = (const float*)d_in[3];  // [1024]
    const float* gamma = (const float*)d_in[4];  // [1024]
    const float* beta  = (const float*)d_in[5];  // [1024]
    float* out = (float*)d_out;                  // [64, 512, 1024]

    // Phase 1: xW + b -> d_out (in-place input for phase 2)
    lnrnn_xw_gemm_f32<<<(BATCH * SEQ) / 16, 512, 0, stream>>>(X, W, bias, out);
    // Phase 2: sequential scan, batch-parallel across 4 workgroups
    lnrnn_recurrent_f32<<<BATCH / 16, 512, 0, stream>>>(U, gamma, beta, out);
}